// ChebConv_24335284699204
// MI455X (gfx1250) — compile-verified
//
#include <hip/hip_runtime.h>
#include <hip/hip_bf16.h>

// ---------------------------------------------------------------------------
// ChebConv: out[b] = sum_k (T_k @ x[b]) @ W[k] + bias
// B=4, N=50000, F_IN=F_OUT=64, K1=4, E=800000   (all fp32)
// ---------------------------------------------------------------------------

typedef float v2f __attribute__((ext_vector_type(2)));
typedef float v8f __attribute__((ext_vector_type(8)));

#define BB   4
#define NN   50000
#define FF   64
#define KK1  4
#define EE   800000

// ---------------------------------------------------------------------------
// out[b,n,f] = bias[f]
// ---------------------------------------------------------------------------
__global__ __launch_bounds__(256) void cheb_init_out(float* __restrict__ out,
                                                     const float* __restrict__ bias,
                                                     long total) {
  long i = (long)blockIdx.x * blockDim.x + threadIdx.x;
  if (i < total) out[i] = bias[i & 63];
}

// ---------------------------------------------------------------------------
// SpMM scatter: support[b, rows[e], f] += vals[e] * x[b, cols[e], f]
// thread = (b, e, f) with f = tid & 63 -> coalesced gather & coalesced atomic
// ---------------------------------------------------------------------------
__global__ __launch_bounds__(256) void cheb_spmm(const float* __restrict__ x,
                                                 const int*   __restrict__ rows,
                                                 const int*   __restrict__ cols,
                                                 const float* __restrict__ vals,
                                                 float*       __restrict__ support) {
  long tid = (long)blockIdx.x * blockDim.x + threadIdx.x;
  long perB = (long)EE * FF;
  if (tid >= (long)BB * perB) return;
  int  b   = (int)(tid / perB);
  long rem = tid - (long)b * perB;
  int  e   = (int)(rem >> 6);
  int  f   = (int)(rem & 63);

  int   r = rows[e];
  int   c = cols[e];
  float v = vals[e];

  float xv = x[((long)b * NN + c) * FF + f];
  unsafeAtomicAdd(&support[((long)b * NN + r) * FF + f], v * xv);
}

// ---------------------------------------------------------------------------
// Dense GEMM via V_WMMA_F32_16X16X4_F32:
//   out[b, m:m+16, :] += support[b, m:m+16, :] @ W    (W = weight[k], 64x64)
// One wave per 16-row strip; 4 x v8f accumulators cover all 64 output cols.
// W is staged in LDS (16 KB) once per block.
// ---------------------------------------------------------------------------
__global__ __launch_bounds__(256) void cheb_wmma_gemm(const float* __restrict__ support,
                                                      const float* __restrict__ W,
                                                      float*       __restrict__ out,
                                                      int nWaves) {
  __shared__ float sW[FF * FF];  // 64x64 fp32 = 16 KB of the 320 KB WGP LDS

  // cooperative load of weight tile into LDS (256 threads x 16 elems)
  #pragma unroll
  for (int i = 0; i < 16; ++i) {
    int idx = threadIdx.x + i * 256;
    sW[idx] = W[idx];
  }
  __syncthreads();

  int wave = blockIdx.x * (blockDim.x >> 5) + (threadIdx.x >> 5);
  if (wave >= nWaves) return;
  int lane = threadIdx.x & 31;

  const int tilesPerBatch = NN >> 4;            // 3125
  int b  = wave / tilesPerBatch;
  int mt = wave - b * tilesPerBatch;

  const float* A = support + ((long)b * NN + (long)mt * 16) * FF;
  float*       O = out     + ((long)b * NN + (long)mt * 16) * FF;

  int half = lane >> 4;       // 0: lanes 0-15, 1: lanes 16-31
  int nl   = lane & 15;       // N index within tile / M index for A rows
  int kb   = half * 2;        // K sub-offset for A/B fragments

  // C initialization: accumulate into existing out (per-k accumulation)
  v8f acc[4];
  #pragma unroll
  for (int nt = 0; nt < 4; ++nt) {
    #pragma unroll
    for (int j = 0; j < 8; ++j) {
      acc[nt][j] = O[(long)(j + 8 * half) * FF + nt * 16 + nl];
    }
  }

  // K loop: F_IN = 64 = 16 steps of k=4
  #pragma unroll
  for (int kk = 0; kk < 16; ++kk) {
    int k0 = kk * 4;

    // A fragment: lane holds (M = nl, K = k0+kb, k0+kb+1) -> contiguous b64 load
    const float* ap = A + (long)nl * FF + (k0 + kb);
    v2f a;
    a.x = ap[0];
    a.y = ap[1];

    #pragma unroll
    for (int nt = 0; nt < 4; ++nt) {
      // B fragment from LDS: lane holds (K = k0+kb, k0+kb+1 ; N = nt*16+nl)
      v2f bfrag;
      bfrag.x = sW[(k0 + kb)     * FF + nt * 16 + nl];
      bfrag.y = sW[(k0 + kb + 1) * FF + nt * 16 + nl];

      acc[nt] = __builtin_amdgcn_wmma_f32_16x16x4_f32(
          /*neg_a=*/false, a, /*neg_b=*/false, bfrag,
          /*c_mod=*/(short)0, acc[nt], /*reuse_a=*/false, /*reuse_b=*/false);
    }
  }

  // store D
  #pragma unroll
  for (int nt = 0; nt < 4; ++nt) {
    #pragma unroll
    for (int j = 0; j < 8; ++j) {
      O[(long)(j + 8 * half) * FF + nt * 16 + nl] = acc[nt][j];
    }
  }
}

// ---------------------------------------------------------------------------
// launcher: init out with bias, then per Chebyshev order k:
//   zero support -> scatter SpMM -> WMMA GEMM accumulate into out
// ---------------------------------------------------------------------------
extern "C" void kernel_launch(void* const* d_in, const int* in_sizes, int n_in,
                              void* d_out, int out_size, void* d_ws, size_t ws_size,
                              hipStream_t stream) {
  const float* x      = (const float*)d_in[0];   // [B, N, F_IN]
  const int*   rows   = (const int*)  d_in[1];   // [K1, E]
  const int*   cols   = (const int*)  d_in[2];   // [K1, E]
  const float* vals   = (const float*)d_in[3];   // [K1, E]
  const float* weight = (const float*)d_in[4];   // [K1, F_IN, F_OUT]
  const float* bias   = (const float*)d_in[5];   // [F_OUT]

  float* out     = (float*)d_out;                // [B, N, F_OUT]
  float* support = (float*)d_ws;                 // [B, N, F_IN] scratch (51.2 MB)

  const long totalOut = (long)BB * NN * FF;      // 12.8M
  cheb_init_out<<<(int)((totalOut + 255) / 256), 256, 0, stream>>>(out, bias, totalOut);

  const long totalSp  = (long)BB * EE * FF;      // 204.8M threads
  const int  spBlocks = (int)((totalSp + 255) / 256);

  const int nWaves    = BB * (NN / 16);          // 12500 strips
  const int gemmBlks  = (nWaves + 7) / 8;        // 8 waves / 256-thread block

  for (int k = 0; k < KK1; ++k) {
    hipMemsetAsync(support, 0, (size_t)BB * NN * FF * sizeof(float), stream);
    cheb_spmm<<<spBlocks, 256, 0, stream>>>(x,
                                            rows + (long)k * EE,
                                            cols + (long)k * EE,
                                            vals + (long)k * EE,
                                            support);
    cheb_wmma_gemm<<<gemmBlks, 256, 0, stream>>>(support,
                                                 weight + (long)k * FF * FF,
                                                 out, nWaves);
  }
}